// SamplesLoss_61383672595026
// MI455X (gfx1250) — compile-verified
//
#include <hip/hip_runtime.h>
#include <math.h>

typedef __attribute__((ext_vector_type(2))) float v2f;
typedef __attribute__((ext_vector_type(8))) float v8f;

#define BATCH 8
#define NPTS 2048
#define SPLITS 4
#define SC (NPTS / SPLITS)          // 512 columns per split
#define WAVES_PER_BLOCK 8
#define ROWS_PER_BLOCK (16 * WAVES_PER_BLOCK)   // 128
#define ROWBLOCKS (NPTS / ROWS_PER_BLOCK)       // 16

// 0.5*acos(x) with the 0.5 folded into the Hastings polynomial.
// Uses raw v_sqrt_f32 (arg in [1e-6,2]: no denorm/negative, ~1ulp is plenty).
__device__ __forceinline__ float half_acos_fast(float x) {
    float ax = __builtin_fabsf(x);
    float p = fmaf(ax, -0.00936465f, 0.03713050f);
    p = fmaf(ax, p, -0.10605720f);
    p = fmaf(ax, p, 0.78536440f);
    float r = __builtin_amdgcn_sqrtf(1.0f - ax) * p;     // 0.5*acos(|x|)
    return (x >= 0.0f) ? r : (1.57079632679f - r);       // 0.5*(pi - acos(|x|))
}

// One softmin pass over C[rows(P), cols(Q)], column-split partial logsumexp.
// wl (staged in LDS) = lw[col] + pot[col] * inv_eps.
// Each wave: 16 rows x 512 cols via 32 V_WMMA_F32_16X16X4_F32 tiles.
__global__ __launch_bounds__(256) void softmin_partial(
    const float* __restrict__ P, const float* __restrict__ Q,
    const float* __restrict__ lw, const float* __restrict__ pot,
    float* __restrict__ pm, float* __restrict__ ps,
    float inv_eps, int same)
{
    __shared__ float sQx[SC], sQy[SC], sQz[SC], sWl[SC];

    int bid = blockIdx.x;
    int b = bid / (ROWBLOCKS * SPLITS);
    int rem = bid % (ROWBLOCKS * SPLITS);
    int rowBlock = rem / SPLITS;
    int split = rem % SPLITS;
    int colBase = split * SC;
    int tid = threadIdx.x;

    for (int i = tid; i < SC; i += 256) {
        int gc = colBase + i;
        const float* q = Q + ((size_t)b * NPTS + gc) * 3;
        sQx[i] = q[0]; sQy[i] = q[1]; sQz[i] = q[2];
        sWl[i] = lw[(size_t)b * NPTS + gc] + pot[(size_t)b * NPTS + gc] * inv_eps;
    }
    __syncthreads();

    int wave = tid >> 5;
    int lane = tid & 31;
    int hi = lane >> 4;                 // which 16-lane half
    int l16 = lane & 15;
    int r0 = rowBlock * ROWS_PER_BLOCK + wave * 16;

    // A fragment (16x4 f32): lanes 0-15 hold K=0,1 = (x,y) of row r0+lane;
    // lanes 16-31 hold K=2,3 = (z, 0) of row r0+lane-16.
    const float* prow = P + ((size_t)b * NPTS + (r0 + l16)) * 3;
    v2f afrag;
    afrag.x = hi ? prow[2] : prow[0];
    afrag.y = hi ? 0.0f : prow[1];

    float m[8], s[8];
#pragma unroll
    for (int i = 0; i < 8; ++i) { m[i] = -1e30f; s[i] = 0.0f; }

    int rbase = r0 + 8 * hi;            // D element i -> global row rbase+i

    for (int ct = 0; ct < SC / 16; ++ct) {
        int cloc = ct * 16 + l16;
        float wl = sWl[cloc];
        // B fragment (4x16 f32), mirrored layout of A over columns
        v2f bfrag;
        bfrag.x = hi ? sQz[cloc] : sQx[cloc];
        bfrag.y = hi ? 0.0f : sQy[cloc];

        v8f acc = {0.f,0.f,0.f,0.f,0.f,0.f,0.f,0.f};
        acc = __builtin_amdgcn_wmma_f32_16x16x4_f32(
            false, afrag, false, bfrag, (short)0, acc, false, false);

        int col_g = colBase + cloc;
#pragma unroll
        for (int i = 0; i < 8; ++i) {
            // WMMA output is finite: single v_med3 clamp, no canonicalize
            float g = __builtin_amdgcn_fmed3f(acc[i], -0.999999f, 0.999999f);
            float c = half_acos_fast(g);                    // 0.5*acos(clip(g))
            if (same && (rbase + i == col_g)) c = 7.0710678e-4f; // 0.5*acos(1-1e-6)
            float t = fmaf(-c, inv_eps, wl);
            float nm = fmaxf(m[i], t);
            s[i] = fmaf(s[i], __expf(m[i] - nm), __expf(t - nm));
            m[i] = nm;
        }
    }

    // butterfly-combine (m,s) across the 16 lanes of each half
#pragma unroll
    for (int k = 1; k < 16; k <<= 1) {
#pragma unroll
        for (int i = 0; i < 8; ++i) {
            float om = __shfl_xor(m[i], k, 32);
            float os = __shfl_xor(s[i], k, 32);
            float nm = fmaxf(m[i], om);
            s[i] = fmaf(s[i], __expf(m[i] - nm), os * __expf(om - nm));
            m[i] = nm;
        }
    }
    if (l16 == 0) {
        size_t base = ((size_t)b * SPLITS + split) * NPTS + rbase;
#pragma unroll
        for (int i = 0; i < 8; ++i) { pm[base + i] = m[i]; ps[base + i] = s[i]; }
    }
}

// merge SPLITS partials, apply -eps*(m+log s) and damping blend
__global__ __launch_bounds__(256) void softmin_combine(
    const float* __restrict__ pm, const float* __restrict__ ps,
    const float* __restrict__ oldv, float* __restrict__ out,
    float eps, float blend)
{
    int idx = blockIdx.x * 256 + threadIdx.x;   // over BATCH*NPTS
    int b = idx / NPTS, r = idx % NPTS;
    float m = -1e30f, s = 0.0f;
#pragma unroll
    for (int k = 0; k < SPLITS; ++k) {
        size_t o = ((size_t)b * SPLITS + k) * NPTS + r;
        float mm = pm[o], ss = ps[o];
        float nm = fmaxf(m, mm);
        s = fmaf(s, __expf(m - nm), ss * __expf(mm - nm));
        m = nm;
    }
    float res = -eps * (m + __logf(s));
    out[idx] = blend * oldv[idx] + (1.0f - blend) * res;
}

__global__ __launch_bounds__(256) void init_logw(
    const float* __restrict__ alpha, const float* __restrict__ beta,
    float* __restrict__ lwa, float* __restrict__ lwb, float* __restrict__ zero)
{
    int idx = blockIdx.x * 256 + threadIdx.x;
    float a = alpha[idx], bt = beta[idx];
    lwa[idx] = (a  > 0.0f) ? __logf(fmaxf(a,  1e-30f)) : -1e5f;
    lwb[idx] = (bt > 0.0f) ? __logf(fmaxf(bt, 1e-30f)) : -1e5f;
    zero[idx] = 0.0f;
}

__global__ __launch_bounds__(256) void loss_kernel(
    const float* __restrict__ alpha, const float* __restrict__ beta,
    const float* __restrict__ fax, const float* __restrict__ fby,
    const float* __restrict__ fay, const float* __restrict__ fbx,
    float* __restrict__ out)
{
    __shared__ float red[256];
    int b = blockIdx.x, tid = threadIdx.x;
    float acc = 0.0f;
    for (int n = tid; n < NPTS; n += 256) {
        size_t i = (size_t)b * NPTS + n;
        acc += alpha[i] * (fbx[i] - fax[i]) + beta[i] * (fay[i] - fby[i]);
    }
    red[tid] = acc; __syncthreads();
    for (int off = 128; off > 0; off >>= 1) {
        if (tid < off) red[tid] += red[tid + off];
        __syncthreads();
    }
    if (tid == 0) out[b] = red[0];
}

extern "C" void kernel_launch(void* const* d_in, const int* in_sizes, int n_in,
                              void* d_out, int out_size, void* d_ws, size_t ws_size,
                              hipStream_t stream)
{
    const float* alpha = (const float*)d_in[0];
    const float* x     = (const float*)d_in[1];
    const float* beta  = (const float*)d_in[2];
    const float* y     = (const float*)d_in[3];
    float* out = (float*)d_out;

    const size_t BN = (size_t)BATCH * NPTS;
    float* w = (float*)d_ws;
    float* zero = w;  w += BN;
    float* lwa  = w;  w += BN;
    float* lwb  = w;  w += BN;
    float* ax[2] = { w, w + BN };  w += 2 * BN;
    float* by[2] = { w, w + BN };  w += 2 * BN;
    float* ay[2] = { w, w + BN };  w += 2 * BN;
    float* bx[2] = { w, w + BN };  w += 2 * BN;
    float* fax = w;  w += BN;
    float* fby = w;  w += BN;
    float* fay = w;  w += BN;
    float* fbx = w;  w += BN;
    float* pm = w;   w += BN * SPLITS;
    float* ps = w;   w += BN * SPLITS;

    // geomloss eps schedule: [d^2] + exp(arange(2ln d, 2ln blur, 2ln 0.5)) + [blur^2]
    const float EPS[8] = {4.0f, 4.0f, 1.0f, 0.25f, 0.0625f,
                          0.015625f, 0.00390625f, 0.0025f};

    dim3 blk(256);
    dim3 gridP(BATCH * ROWBLOCKS * SPLITS);      // 512 blocks
    dim3 gridC((unsigned)(BN / 256));            // 64 blocks

    init_logw<<<gridC, blk, 0, stream>>>(alpha, beta, lwa, lwb, zero);

    auto softmin = [&](const float* P, const float* Q, const float* lw,
                       const float* pot, const float* oldv, float* o,
                       float eps, float blend, int same) {
        softmin_partial<<<gridP, blk, 0, stream>>>(P, Q, lw, pot, pm, ps,
                                                   1.0f / eps, same);
        softmin_combine<<<gridC, blk, 0, stream>>>(pm, ps, oldv, o, eps, blend);
    };

    // ---- init at eps = EPS[0], pot = 0, no blend ----
    softmin(x, x, lwa, zero, zero, ax[0], EPS[0], 0.0f, 1);  // a_x
    softmin(y, y, lwb, zero, zero, by[0], EPS[0], 0.0f, 1);  // b_y
    softmin(y, x, lwa, zero, zero, ay[0], EPS[0], 0.0f, 0);  // a_y (C_yx)
    softmin(x, y, lwb, zero, zero, bx[0], EPS[0], 0.0f, 0);  // b_x (C_xy)

    // ---- annealed symmetrized loop, damping 0.5, ping-pong buffers ----
    for (int it = 0; it < 8; ++it) {
        float eps = EPS[it];
        int c = it & 1, nx = 1 - c;
        softmin(x, x, lwa, ax[c], ax[c], ax[nx], eps, 0.5f, 1);  // at_x
        softmin(y, y, lwb, by[c], by[c], by[nx], eps, 0.5f, 1);  // bt_y
        softmin(y, x, lwa, bx[c], ay[c], ay[nx], eps, 0.5f, 0);  // at_y uses b_x
        softmin(x, y, lwb, ay[c], bx[c], bx[nx], eps, 0.5f, 0);  // bt_x uses a_y
    }
    // after 8 iterations the live set is index 0

    // ---- final extrapolation at eps = blur^2 ----
    float ef = EPS[7];
    softmin(x, x, lwa, ax[0], zero, fax, ef, 0.0f, 1);
    softmin(y, y, lwb, by[0], zero, fby, ef, 0.0f, 1);
    softmin(y, x, lwa, bx[0], zero, fay, ef, 0.0f, 0);
    softmin(x, y, lwb, ay[0], zero, fbx, ef, 0.0f, 0);

    loss_kernel<<<dim3(BATCH), blk, 0, stream>>>(alpha, beta, fax, fby, fay, fbx, out);
}